// WeightedAggregator_multi_89489938580184
// MI455X (gfx1250) — compile-verified
//
#include <hip/hip_runtime.h>

// Weighted multi-mask neighbor aggregation for MI455X (gfx1250, wave32).
// out[b, j*128+d] = sum_k Wn[j][k] * embed[idx[b][k]][d],  Wn[j] = w^(0.5j) row-normalized.
// One wave32 per batch row; per-row GEMM (4x32)x(32x128) mapped onto
// V_WMMA_F32_16X16X4_F32 (full f32 precision; kernel is HBM-gather bound anyway).

typedef float v2f __attribute__((ext_vector_type(2)));
typedef float v8f __attribute__((ext_vector_type(8)));

#define D_FEAT 128
#define K_NEIGH 32
#define N_MASKS 4
#define WAVES_PER_BLOCK 8

__global__ __launch_bounds__(256) void wagg_kernel(
    const float* __restrict__ embed,
    const int*   __restrict__ neigh_idx,
    const float* __restrict__ neigh_w,
    float*       __restrict__ out,
    int batch)
{
    // Per-wave LDS: 16x32 f32 A-tile (rows 4..15 zero) + 32 neighbor indices.
    __shared__ float lds_wn [WAVES_PER_BLOCK][16 * K_NEIGH];
    __shared__ int   lds_idx[WAVES_PER_BLOCK][K_NEIGH];

    const int lane = threadIdx.x & 31;
    const int wave = threadIdx.x >> 5;
    const int b    = blockIdx.x * WAVES_PER_BLOCK + wave;
    if (b >= batch) return;   // whole wave exits together -> EXEC stays all-1 for WMMA

    // ---- Step 1: per-lane weight, the 4 mask values, wave reductions ----
    const float w  = neigh_w[(size_t)b * K_NEIGH + lane];
    const float r  = __builtin_sqrtf(w);
    const float m1 = r;          // w^0.5
    const float m2 = w;          // w^1.0
    const float m3 = w * r;      // w^1.5

    float s1 = m1, s2 = m2, s3 = m3;
    #pragma unroll
    for (int off = 16; off > 0; off >>= 1) {
        s1 += __shfl_xor(s1, off, 32);
        s2 += __shfl_xor(s2, off, 32);
        s3 += __shfl_xor(s3, off, 32);
    }

    float* wnrow = lds_wn[wave];
    wnrow[0 * K_NEIGH + lane] = 1.0f / (float)K_NEIGH;  // sum of w^0 over 32 lanes is exactly 32
    wnrow[1 * K_NEIGH + lane] = m1 / s1;
    wnrow[2 * K_NEIGH + lane] = m2 / s2;
    wnrow[3 * K_NEIGH + lane] = m3 / s3;
    #pragma unroll
    for (int zr = N_MASKS; zr < 16; ++zr)
        wnrow[zr * K_NEIGH + lane] = 0.0f;              // zero-pad A rows 4..15

    lds_idx[wave][lane] = neigh_idx[(size_t)b * K_NEIGH + lane];

    __syncthreads();  // LDS is wave-private but a block barrier is cheap & safe

    // ---- Step 2: preload A fragments for all 8 K-chunks from LDS ----
    // A 16x4 f32 layout: VGPR i, half h (lane>>4) holds K = kc + 2h + i ; M = lane&15.
    const int half = lane >> 4;
    const int m    = lane & 15;

    v2f afrag[8];
    #pragma unroll
    for (int c = 0; c < 8; ++c) {
        const int k0 = c * 4 + 2 * half;
        afrag[c][0] = wnrow[m * K_NEIGH + k0];
        afrag[c][1] = wnrow[m * K_NEIGH + k0 + 1];
    }

    const int* idxrow = lds_idx[wave];
    float* ob = out + (size_t)b * (N_MASKS * D_FEAT);

    // ---- Step 3: 8 d-tiles x 8 k-chunks of v_wmma_f32_16x16x4_f32 ----
    // B 4x16 f32 layout mirrors A: VGPR i, half h holds row K = kc + 2h + i, col = lane&15.
    for (int t = 0; t < 8; ++t) {
        const int d0 = t * 16;
        v8f acc = {};   // C = 0

        #pragma unroll
        for (int c = 0; c < 8; ++c) {
            const int k0   = c * 4 + 2 * half;
            const int row0 = idxrow[k0];
            const int row1 = idxrow[k0 + 1];
            v2f bfrag;
            bfrag[0] = embed[(size_t)row0 * D_FEAT + d0 + m];
            bfrag[1] = embed[(size_t)row1 * D_FEAT + d0 + m];
            // (neg_a, A, neg_b, B, c_mod, C, reuse_a, reuse_b)
            acc = __builtin_amdgcn_wmma_f32_16x16x4_f32(
                false, afrag[c], false, bfrag, (short)0, acc, false, false);
        }

        // D layout: VGPR j, lanes 0-15 = row M=j (cols = lane). Rows 0..3 are the 4 masks.
        if (half == 0) {
            ob[0 * D_FEAT + d0 + m] = acc[0];
            ob[1 * D_FEAT + d0 + m] = acc[1];
            ob[2 * D_FEAT + d0 + m] = acc[2];
            ob[3 * D_FEAT + d0 + m] = acc[3];
        }
    }
}

extern "C" void kernel_launch(void* const* d_in, const int* in_sizes, int n_in,
                              void* d_out, int out_size, void* d_ws, size_t ws_size,
                              hipStream_t stream) {
    const float* embed = (const float*)d_in[0];
    const int*   nidx  = (const int*)  d_in[1];
    const float* nw    = (const float*)d_in[2];
    float*       out   = (float*)d_out;

    const int batch  = in_sizes[1] / K_NEIGH;   // 16384
    const int blocks = (batch + WAVES_PER_BLOCK - 1) / WAVES_PER_BLOCK;

    hipLaunchKernelGGL(wagg_kernel, dim3(blocks), dim3(256), 0, stream,
                       embed, nidx, nw, out, batch);
}